// EventClassifierNet_70858370449746
// MI455X (gfx1250) — compile-verified
//
#include <hip/hip_runtime.h>
#include <math.h>

typedef __attribute__((ext_vector_type(2))) float v2f;
typedef __attribute__((ext_vector_type(8))) float v8f;

#define WAVES 8
constexpr int IN_F = 14, H = 24, T = 25, NC = 4, WH = 12;
constexpr int G  = 4 * H;       // 96 gates
constexpr int NT = G / 16;      // 6 N-tiles of 16
constexpr int KX = 4;           // ceil(14/4) K-slices for x GEMM (padded to 16)
constexpr int KH = H / 4;       // 6 K-slices for h GEMM

// Raw-HW transcendentals: v_exp_f32 + v_rcp_f32 (TRANS ops, co-execute with
// the matrix pipe; ~1 ulp, fine for sigmoid/tanh arguments).
__device__ __forceinline__ float fsig(float x) {
  float e = __builtin_amdgcn_exp2f(x * -1.4426950408889634f);   // exp(-x)
  return __builtin_amdgcn_rcpf(1.f + e);
}
__device__ __forceinline__ float ftanh(float x) {
  float t = __builtin_amdgcn_exp2f(-2.8853900817779268f * fabsf(x)); // exp(-2|x|)
  float r = (1.f - t) * __builtin_amdgcn_rcpf(1.f + t);
  return copysignf(r, x);
}

#define WAVE_FENCE() asm volatile("" ::: "memory")

__global__ __launch_bounds__(32 * WAVES) void bilstm_center_kernel(
    const float* __restrict__ x,
    const float* __restrict__ w_ih_f, const float* __restrict__ w_hh_f,
    const float* __restrict__ b_ih_f, const float* __restrict__ b_hh_f,
    const float* __restrict__ w_ih_b, const float* __restrict__ w_hh_b,
    const float* __restrict__ b_ih_b, const float* __restrict__ b_hh_b,
    const float* __restrict__ head_w, const float* __restrict__ head_b,
    float* __restrict__ out, int B)
{
  // weight fragments (block-shared, written once)
  __shared__ float s_wx[2][NT * KX * 64];   // x-GEMM B fragments   12 KB
  __shared__ float s_wh[2][NT * KH * 64];   // h-GEMM B fragments   18 KB
  __shared__ float s_bias[2][G];
  __shared__ float s_headw[NC * 2 * H];
  __shared__ float s_headb[NC];
  // per-wave staging
  __shared__ float s_gates[WAVES][16 * G];  // 48 KB
  __shared__ float s_h[WAVES][16 * H];      // 12 KB
  __shared__ float s_hf[WAVES][16 * H];     // 12 KB

  const int tid = threadIdx.x;

  // ---- cooperative weight swizzle into exact WMMA B-fragment layout ----
  // B fragment for (nt,kk): slot = lane*2+e holds B[k][n] with
  //   n = nt*16 + (lane&15),  k = kk*4 + (lane>>4)*2 + e   (B = W^T, so B[k][n]=W[n][k])
  for (int i = tid; i < 2 * NT * KX * 64; i += blockDim.x) {
    int d = i / (NT * KX * 64);
    int r = i % (NT * KX * 64);
    int frag = r >> 6, slot = r & 63;
    int nt = frag / KX, kk = frag % KX;
    int ln = slot >> 1, e = slot & 1;
    int n = nt * 16 + (ln & 15);
    int k = kk * 4 + ((ln >> 4) << 1) + e;
    const float* w = d ? w_ih_b : w_ih_f;
    s_wx[d][r] = (k < IN_F) ? w[n * IN_F + k] : 0.f;
  }
  for (int i = tid; i < 2 * NT * KH * 64; i += blockDim.x) {
    int d = i / (NT * KH * 64);
    int r = i % (NT * KH * 64);
    int frag = r >> 6, slot = r & 63;
    int nt = frag / KH, kk = frag % KH;
    int ln = slot >> 1, e = slot & 1;
    int n = nt * 16 + (ln & 15);
    int k = kk * 4 + ((ln >> 4) << 1) + e;
    const float* w = d ? w_hh_b : w_hh_f;
    s_wh[d][r] = w[n * H + k];
  }
  for (int i = tid; i < 2 * G; i += blockDim.x) {
    int d = i / G, n = i % G;
    s_bias[d][n] = d ? (b_ih_b[n] + b_hh_b[n]) : (b_ih_f[n] + b_hh_f[n]);
  }
  for (int i = tid; i < NC * 2 * H; i += blockDim.x) s_headw[i] = head_w[i];
  if (tid < NC) s_headb[tid] = head_b[tid];
  __syncthreads();

  const int lane = tid & 31;
  const int w    = tid >> 5;
  const int M    = lane & 15;   // batch row within tile (A/D row, D col)
  const int hi   = lane >> 4;
  const int b0   = (blockIdx.x * WAVES + w) * 16;
  if (b0 >= B) return;

  float* gatesL = &s_gates[w][0];
  float* hL     = &s_h[w][0];

  // per-lane bias splats: bias for gate n = nt*16 + M
  float bx[2][NT];
  #pragma unroll
  for (int d = 0; d < 2; ++d)
    #pragma unroll
    for (int nt = 0; nt < NT; ++nt) bx[d][nt] = s_bias[d][nt * 16 + M];

  float cst[12];
  const float* xbase = x + (size_t)b0 * T * IN_F;

  for (int d = 0; d < 2; ++d) {
    #pragma unroll
    for (int u = 0; u < 12; ++u) cst[u] = 0.f;
    for (int e = lane; e < 16 * H; e += 32) hL[e] = 0.f;
    WAVE_FENCE();

    for (int step = 0; step <= WH; ++step) {
      const int t = d ? (T - 1 - step) : step;
      const float* xr = xbase + ((size_t)M * T + t) * IN_F;

      // A fragments for x GEMM: lane holds x[M][k], x[M][k+1], k = 4kk+2*hi
      v2f ax[KX];
      #pragma unroll
      for (int kk = 0; kk < KX; ++kk) {
        const int ks = kk * 4 + hi * 2;
        if (ks + 1 < IN_F) ax[kk] = *(const v2f*)(xr + ks);
        else               ax[kk] = (v2f){0.f, 0.f};   // K padding 14..15
      }
      // A fragments for h GEMM from LDS
      v2f ah[KH];
      #pragma unroll
      for (int kk = 0; kk < KH; ++kk)
        ah[kk] = *(const v2f*)&hL[M * H + kk * 4 + hi * 2];

      #pragma unroll
      for (int nt = 0; nt < NT; ++nt) {
        const float bb = bx[d][nt];
        v8f acc;
        #pragma unroll
        for (int r = 0; r < 8; ++r) acc[r] = bb;
        #pragma unroll
        for (int kk = 0; kk < KX; ++kk) {
          v2f bfrag = *(const v2f*)&s_wx[d][(nt * KX + kk) * 64 + lane * 2];
          acc = __builtin_amdgcn_wmma_f32_16x16x4_f32(
              false, ax[kk], false, bfrag, (short)0, acc, false, false);
        }
        #pragma unroll
        for (int kk = 0; kk < KH; ++kk) {
          v2f bfrag = *(const v2f*)&s_wh[d][(nt * KH + kk) * 64 + lane * 2];
          acc = __builtin_amdgcn_wmma_f32_16x16x4_f32(
              false, ah[kk], false, bfrag, (short)0, acc, false, false);
        }
        // D layout: lane L, reg r -> row M=r+8*hi, col N = nt*16 + (L&15)
        #pragma unroll
        for (int r = 0; r < 8; ++r)
          gatesL[(r + 8 * hi) * G + nt * 16 + M] = acc[r];
      }
      WAVE_FENCE();

      // LSTM cell update: 384 (m,j) cells, 12 per lane; c in registers
      #pragma unroll
      for (int u = 0; u < 12; ++u) {
        const int e = u * 32 + lane;
        const int m = e / H, j = e - m * H;
        const float gi = gatesL[m * G + j];
        const float gf = gatesL[m * G + 24 + j];
        const float gg = gatesL[m * G + 48 + j];
        const float go = gatesL[m * G + 72 + j];
        const float iv = fsig(gi), fv = fsig(gf), ov = fsig(go), gv = ftanh(gg);
        const float cc = fv * cst[u] + iv * gv;
        cst[u] = cc;
        hL[e] = ov * ftanh(cc);   // e == m*H + j
      }
      WAVE_FENCE();
    }

    if (d == 0) {  // save forward h(t=12)
      for (int e = lane; e < 16 * H; e += 32) s_hf[w][e] = hL[e];
      WAVE_FENCE();
    }
  }

  // head: out[m][cls] = b[cls] + hf . Wf + hb . Wb   (64 outputs, 2 per lane)
  #pragma unroll
  for (int p = lane; p < 64; p += 32) {
    const int m = p >> 2, cls = p & 3;
    float acc = s_headb[cls];
    #pragma unroll
    for (int j = 0; j < H; ++j) {
      acc += s_hf[w][m * H + j] * s_headw[cls * 2 * H + j];
      acc += hL[m * H + j]      * s_headw[cls * 2 * H + H + j];
    }
    out[(size_t)(b0 + m) * NC + cls] = acc;
  }
}

extern "C" void kernel_launch(void* const* d_in, const int* in_sizes, int n_in,
                              void* d_out, int out_size, void* d_ws, size_t ws_size,
                              hipStream_t stream) {
  (void)n_in; (void)d_ws; (void)ws_size; (void)out_size;
  const float* x      = (const float*)d_in[0];
  const float* w_ih_f = (const float*)d_in[1];
  const float* w_hh_f = (const float*)d_in[2];
  const float* b_ih_f = (const float*)d_in[3];
  const float* b_hh_f = (const float*)d_in[4];
  const float* w_ih_b = (const float*)d_in[5];
  const float* w_hh_b = (const float*)d_in[6];
  const float* b_ih_b = (const float*)d_in[7];
  const float* b_hh_b = (const float*)d_in[8];
  const float* head_w = (const float*)d_in[9];
  const float* head_b = (const float*)d_in[10];
  float* out = (float*)d_out;

  const int B = in_sizes[0] / (T * IN_F);            // 65536
  const int tilesPerBlock = WAVES;                   // 16 rows per wave
  const int grid = (B + 16 * tilesPerBlock - 1) / (16 * tilesPerBlock);
  bilstm_center_kernel<<<grid, 32 * WAVES, 0, stream>>>(
      x, w_ih_f, w_hh_f, b_ih_f, b_hh_f,
      w_ih_b, w_hh_b, b_ih_b, b_hh_b, head_w, head_b, out, B);
}